// RBFSetConv_33792802685155
// MI455X (gfx1250) — compile-verified
//
#include <hip/hip_runtime.h>
#include <math.h>

// ---------------------------------------------------------------------------
// RBF set-convolution onto a regular 128x128 grid, restructured as GEMM.
//
//   exp(-|x-g|^2 / 2s^2) = exp(-dx^2/2s^2) * exp(-dy^2/2s^2)   (separable)
//
//   density[j,i]  = sum_n (mask*WY)[n,j]   * WX[n,i]
//   weighted[j,i] = sum_n (mask*y*WY)[n,j] * WX[n,i]
//
// -> two 128x128x4096 f32 GEMMs per batch (shared B operand), mapped onto
//    V_WMMA_F32_16X16X4_F32 (wave32, CDNA5).
// ---------------------------------------------------------------------------

#define GRID_RES   128
#define NPTS       4096
#define NBATCH     4
#define NC         64          // N-chunk staged in LDS per iteration
#define INV2SIG2   50.0f       // 1 / (2 * 0.1^2)
#define EPS_NORM   1e-5f

typedef float v8f __attribute__((ext_vector_type(8)));
typedef float v2f __attribute__((ext_vector_type(2)));

__global__ __launch_bounds__(256) void
rbf_setconv_wmma_kernel(const float* __restrict__ x_c,          // (B,N,2)
                        const float* __restrict__ y_c,          // (B,N,1)
                        const float* __restrict__ grid_points,  // (2,M)
                        const unsigned char* __restrict__ mask, // (B,N) bool
                        float* __restrict__ out)                // (B,2,128,128)
{
    // LDS staging (padded strides to avoid bank conflicts on the WMMA feeds)
    __shared__ float sWX [NC][72];   // WX[n][i]        : B operand (K x 128)
    __shared__ float sWY0[16][66];   // (mask*WY)^T[j][n]   : A operand, density
    __shared__ float sWY1[16][66];   // (mask*y*WY)^T[j][n] : A operand, weighted
    __shared__ float cxs[GRID_RES];  // grid x coords
    __shared__ float cys[GRID_RES];  // grid y coords

    const int t    = threadIdx.x;
    const int blk  = blockIdx.x;
    const int b    = blk >> 3;          // batch
    const int j0   = (blk & 7) << 4;    // 16-row output slab
    const int wave = t >> 5;
    const int lane = t & 31;
    const int i0   = wave << 4;         // this wave's 16-col output tile

    // grid coords: m = j*128 + i ; gp[0][m] = coords[i], gp[1][m] = coords[j]
    if (t < GRID_RES) {
        cxs[t] = grid_points[t];
        cys[t] = grid_points[GRID_RES * GRID_RES + t * GRID_RES];
    }

    v8f accD = {};   // density accumulator   (16x16 f32 C/D tile)
    v8f accW = {};   // weighted accumulator

    // WMMA f32 16x16x4 operand lane mapping:
    //   A (16x4): lane%16 = M row; lanes 0-15 hold K = k0,k0+1 ; lanes 16-31 K = k0+2,k0+3
    //   B (4x16): lane%16 = N col; same K split across lane halves
    const int rowA = lane & 15;
    const int kOff = (lane >> 4) << 1;
    const int colB = i0 + (lane & 15);

    for (int nc = 0; nc < NPTS; nc += NC) {
        __syncthreads();   // previous chunk's GEMM reads done (also covers coord init)

        // ---- stage WX[n][i] for this chunk: 64 rows x 128 cols, 256 threads ----
        {
            const int n = t >> 2;
            const int q = t & 3;
            const size_t base = (size_t)b * NPTS + nc + n;
            const float xv = x_c[(base << 1) + 0];
            #pragma unroll
            for (int ii = 0; ii < 32; ++ii) {
                const int col = (q << 5) + ii;
                const float d = xv - cxs[col];
                sWX[n][col] = expf(-INV2SIG2 * d * d);
            }
        }

        // ---- stage (mask*WY)^T and (mask*y*WY)^T for this slab's 16 j rows ----
        if (t < NC) {
            const int n = t;
            const size_t base = (size_t)b * NPTS + nc + n;
            const float xv = x_c[(base << 1) + 1];
            const float yv = y_c[base];
            const float mk = mask[base] ? 1.0f : 0.0f;
            #pragma unroll
            for (int j = 0; j < 16; ++j) {
                const float d = xv - cys[j0 + j];
                const float w = expf(-INV2SIG2 * d * d) * mk;
                sWY0[j][n] = w;
                sWY1[j][n] = w * yv;
            }
        }
        __syncthreads();

        // ---- K-loop: rank-4 WMMA updates over the chunk ----
        for (int k0 = 0; k0 < NC; k0 += 4) {
            const int ca = k0 + kOff;
            const v2f a0 = *(const v2f*)&sWY0[rowA][ca];   // contiguous K pair
            const v2f a1 = *(const v2f*)&sWY1[rowA][ca];
            v2f bv;
            bv.x = sWX[ca    ][colB];
            bv.y = sWX[ca + 1][colB];
            accD = __builtin_amdgcn_wmma_f32_16x16x4_f32(
                       false, a0, false, bv, (short)0, accD, false, false);
            accW = __builtin_amdgcn_wmma_f32_16x16x4_f32(
                       false, a1, false, bv, (short)0, accW, false, false);
        }
    }

    // ---- epilogue: C/D layout -> global; normalize channel 1 in-register ----
    // lane%16 = N col; VGPR v holds M = v + 8*(lane/16)
    float* ob = out + (size_t)b * 2 * GRID_RES * GRID_RES;
    const int mHi = (lane >> 4) << 3;
    #pragma unroll
    for (int v = 0; v < 8; ++v) {
        const int jj = j0 + v + mHi;
        const int m  = jj * GRID_RES + colB;
        const float d = accD[v];
        const float w = accW[v];
        ob[m] = d;                                          // channel 0: density
        ob[GRID_RES * GRID_RES + m] = w / (d + EPS_NORM);   // channel 1: normalized
    }
}

extern "C" void kernel_launch(void* const* d_in, const int* in_sizes, int n_in,
                              void* d_out, int out_size, void* d_ws, size_t ws_size,
                              hipStream_t stream) {
    const float*         x_c  = (const float*)d_in[0];          // (4,4096,2)
    const float*         y_c  = (const float*)d_in[1];          // (4,4096,1)
    const float*         gp   = (const float*)d_in[2];          // (2,16384)
    const unsigned char* mask = (const unsigned char*)d_in[3];  // (4,4096) bool
    float*               out  = (float*)d_out;                  // (4,2,128,128)

    (void)in_sizes; (void)n_in; (void)out_size; (void)d_ws; (void)ws_size;

    dim3 grid(NBATCH * 8);   // batch x 8 row-slabs
    dim3 block(256);         // 8 waves of 32
    rbf_setconv_wmma_kernel<<<grid, block, 0, stream>>>(x_c, y_c, gp, mask, out);
}